// Refine_moduleGNN_4209067950245
// MI455X (gfx1250) — compile-verified
//
#include <hip/hip_runtime.h>

typedef __attribute__((ext_vector_type(2))) float v2f;
typedef __attribute__((ext_vector_type(8))) float v8f;

#define NPTS  512
#define BSZ   16
#define CDIM  256
#define HDIM  128
#define GDIM_ 64
#define KNN_K 20

// ---------------------------------------------------------------------------
// CDNA5 async global->LDS copy (16 bytes per lane) + wait, via inline asm.
// LDS address comes from the vdst VGPR plus offset; LDS offset = low 32 bits of flat shared
// address (ISA 10.2: LDS aperture keeps the offset in addr[31:0]).
// ---------------------------------------------------------------------------
__device__ __forceinline__ void async_copy_b128(const float* lds_dst,
                                                const float* gsrc) {
  unsigned lds_off = (unsigned)(size_t)lds_dst;
  unsigned long long ga = (unsigned long long)(size_t)gsrc;
  asm volatile("global_load_async_to_lds_b128 %0, %1, off"
               :: "v"(lds_off), "v"(ga)
               : "memory");
}
__device__ __forceinline__ void wait_async0() {
  asm volatile("s_wait_asynccnt 0" ::: "memory");
}

// ---------------------------------------------------------------------------
// Kernel 1: gathered-conv as WMMA GEMM (register-pipelined).
// Rows r = ((b*512+n)*4 + corner), M = 32768. K = 4096 (ch*16 + ky*4 + kx).
// N = 64 outputs. A gathered on the fly from img with zero padding; B is
// conv_w reinterpreted as K x 64 (strided). Epilogue adds conv bias, applies
// roi mask, scatters into local320[(b*512+n)*320 + corner*64 + o].
// ---------------------------------------------------------------------------
__global__ __launch_bounds__(256) void conv_gather_wmma(
    const float* __restrict__ img, const int* __restrict__ px,
    const int* __restrict__ py, const float* __restrict__ w,
    const float* __restrict__ bias, const float* __restrict__ roi,
    float* __restrict__ outL) {
  __shared__ __align__(16) float As[64 * 68];
  __shared__ __align__(16) float Bs[64 * 68];
  const int tid = threadIdx.x;
  const int lane = tid & 31;
  const int wv = tid >> 5;
  const int rowbase = blockIdx.x * 64;

  // Per-thread A-load coordinates: thread handles row am, 16 consecutive k.
  const int am = tid >> 2;
  const int akb = (tid & 3) << 4;
  const int r = rowbase + am;
  const int ab = r >> 11;            // / (512*4)
  const int arem = r & 2047;
  const int an = arem >> 2;
  const int acorner = arem & 3;
  const int oy = 2 * py[ab * NPTS + an] + ((acorner & 1) ? 4 : 0);
  const int ox = 2 * px[ab * NPTS + an] + ((acorner & 2) ? 4 : 0);

  const int bkk = tid >> 2;
  const int bob = (tid & 3) << 4;

  const int m0 = (wv >> 1) * 16;
  const int n0 = (wv & 1) * 32;
  const int lrow = lane & 15;
  const int khalf = (lane >> 4) << 1;    // 0 for lanes 0-15, 2 for 16-31
  v8f acc0 = {};
  v8f acc1 = {};

  float areg[16], breg[16];
  auto load_tile = [&](int kt) {
    const int ch = (kt + akb) >> 4;
    const float* ip = img + (size_t)(ab * CDIM + ch) * HDIM * HDIM;
#pragma unroll
    for (int j = 0; j < 16; ++j) {
      const int ky = j >> 2, kx = j & 3;
      const int iy = oy - 3 + ky, ix = ox - 3 + kx;
      float v = 0.f;
      if (iy >= 0 && iy < HDIM && ix >= 0 && ix < HDIM) v = ip[iy * HDIM + ix];
      areg[j] = v;
    }
#pragma unroll
    for (int j = 0; j < 16; ++j)
      breg[j] = w[(size_t)(bob + j) * 4096 + kt + bkk];
  };

  load_tile(0);
  for (int kt = 0; kt < 4096; kt += 64) {
    __syncthreads();   // LDS free (previous tile fully consumed)
#pragma unroll
    for (int j = 0; j < 16; ++j) As[am * 68 + akb + j] = areg[j];
#pragma unroll
    for (int j = 0; j < 16; ++j) Bs[bkk * 68 + bob + j] = breg[j];
    __syncthreads();
    if (kt + 64 < 4096) load_tile(kt + 64);  // overlaps with WMMA below
#pragma unroll
    for (int kk = 0; kk < 64; kk += 4) {
      v2f a, vb0, vb1;
      const float* ar = &As[(m0 + lrow) * 68 + kk + khalf];
      a.x = ar[0];
      a.y = ar[1];
      vb0.x = Bs[(kk + khalf) * 68 + n0 + lrow];
      vb0.y = Bs[(kk + khalf + 1) * 68 + n0 + lrow];
      vb1.x = Bs[(kk + khalf) * 68 + n0 + 16 + lrow];
      vb1.y = Bs[(kk + khalf + 1) * 68 + n0 + 16 + lrow];
      acc0 = __builtin_amdgcn_wmma_f32_16x16x4_f32(false, a, false, vb0,
                                                   (short)0, acc0, false, false);
      acc1 = __builtin_amdgcn_wmma_f32_16x16x4_f32(false, a, false, vb1,
                                                   (short)0, acc1, false, false);
    }
  }
  // Epilogue: C VGPR r -> (M = m0+r (+8 for lanes>=16), N = n0 + lane&15)
#pragma unroll
  for (int rr = 0; rr < 8; ++rr) {
    const int mrow = m0 + rr + ((lane >> 4) << 3);
    const int gr = rowbase + mrow;
    const int gb = gr >> 11;
    const int grm = gr & 2047;
    const int gn = grm >> 2;
    const int gcorner = grm & 3;
    const float msk = roi[gb * NPTS + gn];
    float* dst = outL + (size_t)(gb * NPTS + gn) * 320 + gcorner * 64;
    const int o0 = n0 + lrow;
    const int o1 = o0 + 16;
    dst[o0] = (acc0[rr] + bias[o0]) * msk;
    dst[o1] = (acc1[rr] + bias[o1]) * msk;
  }
}

// ---------------------------------------------------------------------------
// Kernel 2: concat graph_feat (B,64,N) into local320[..., 256 + g]
// ---------------------------------------------------------------------------
__global__ void graph_concat(const float* __restrict__ gf, float* __restrict__ outL) {
  const int e = blockIdx.x * blockDim.x + threadIdx.x;
  if (e >= BSZ * NPTS * GDIM_) return;
  const int g = e & 63;
  const int n = (e >> 6) & 511;
  const int b = e >> 15;
  outL[(size_t)(b * NPTS + n) * 320 + 256 + g] = gf[(size_t)(b * GDIM_ + g) * NPTS + n];
}

// ---------------------------------------------------------------------------
// Kernel 3: EdgeConv weight prep: Wu[c][o] = W[o][c], Wv[c][o] = W[o][256+c]-W[o][c]
// ---------------------------------------------------------------------------
__global__ void edge_weight_prep(const float* __restrict__ gw,
                                 float* __restrict__ Wu, float* __restrict__ Wv) {
  const int e = blockIdx.x * blockDim.x + threadIdx.x;
  if (e >= CDIM * CDIM) return;
  const int o = e & 255;
  const int c = e >> 8;
  const float wa = gw[(size_t)o * 512 + c];
  const float wb = gw[(size_t)o * 512 + 256 + c];
  Wu[e] = wa;
  Wv[e] = wb - wa;
}

// ---------------------------------------------------------------------------
// Kernel 4: generic WMMA GEMM, C = lrelu(A(MxK) * B(KxN) + bias, slope)
// BM=64, BN=32, Kc=32 tiles; 8 wave32 per block; one 16x16 f32 WMMA acc/wave.
// Double-buffered LDS filled by CDNA5 async global->LDS DMA: the async copy
// for tile t+1 runs concurrently with the WMMA loop over tile t.
// ---------------------------------------------------------------------------
__global__ __launch_bounds__(256) void gemm_bias_lrelu(
    const float* __restrict__ A, const float* __restrict__ Bm,
    const float* __restrict__ bias, float* __restrict__ C, int K, int Nld,
    float slope) {
  __shared__ __align__(16) float As[2][64 * 36];
  __shared__ __align__(16) float Bs[2][32 * 36];
  const int tid = threadIdx.x;
  const int lane = tid & 31;
  const int wv = tid >> 5;
  const int rowbase = blockIdx.x * 64;
  const int colbase = blockIdx.y * 32;

  const int am = tid >> 2;          // A: 8 floats -> two b128 async copies
  const int akb = (tid & 3) << 3;
  const int bkk = tid >> 3;         // B: 4 floats -> one b128 async copy
  const int bob = (tid & 7) << 2;

  const int m0 = (wv >> 1) * 16;
  const int n0 = (wv & 1) * 16;
  const int lrow = lane & 15;
  const int khalf = (lane >> 4) << 1;
  v8f acc = {};

  auto stage = [&](int buf, int kt) {
    const float* ag = A + (size_t)(rowbase + am) * K + kt + akb;
    const float* ad = &As[buf][am * 36 + akb];
    async_copy_b128(ad, ag);
    async_copy_b128(ad + 4, ag + 4);
    const float* bg = Bm + (size_t)(kt + bkk) * Nld + colbase + bob;
    async_copy_b128(&Bs[buf][bkk * 36 + bob], bg);
  };

  stage(0, 0);
  int cur = 0;
  for (int kt = 0; kt < K; kt += 32) {
    wait_async0();       // tile `cur` landed in LDS (per-wave)
    __syncthreads();     // all waves: loads visible, prev compute finished
    if (kt + 32 < K) stage(cur ^ 1, kt + 32);  // DMA next tile during WMMAs
#pragma unroll
    for (int kk = 0; kk < 32; kk += 4) {
      v2f a, vb;
      const float* ar = &As[cur][(m0 + lrow) * 36 + kk + khalf];
      a.x = ar[0];
      a.y = ar[1];
      vb.x = Bs[cur][(kk + khalf) * 36 + n0 + lrow];
      vb.y = Bs[cur][(kk + khalf + 1) * 36 + n0 + lrow];
      acc = __builtin_amdgcn_wmma_f32_16x16x4_f32(false, a, false, vb, (short)0,
                                                  acc, false, false);
    }
    cur ^= 1;
  }
  const int gcol = colbase + n0 + lrow;
  const float bv = bias ? bias[gcol] : 0.f;
#pragma unroll
  for (int rr = 0; rr < 8; ++rr) {
    const int grow = rowbase + m0 + rr + ((lane >> 4) << 3);
    const float v = acc[rr] + bv;
    C[(size_t)grow * Nld + gcol] = (v >= 0.f) ? v : slope * v;
  }
}

// ---------------------------------------------------------------------------
// Kernel 5: EdgeConv gather + BN + lrelu(0.2) + max over k neighbors.
// y[b,n,k,o] = U[b,knn[n,k],o] + V[b,n,o]; one block per (b,n), thread = o.
// ---------------------------------------------------------------------------
__global__ __launch_bounds__(256) void edge_gather_max(
    const float* __restrict__ U, const float* __restrict__ V,
    const int* __restrict__ knn, const float* __restrict__ gamma,
    const float* __restrict__ beta, const float* __restrict__ mean,
    const float* __restrict__ var, float* __restrict__ Out) {
  const int bn = blockIdx.x;
  const int n = bn & 511;
  const int b = bn >> 9;
  const int o = threadIdx.x;
  const float vv = V[(size_t)bn * CDIM + o];
  const float sc = gamma[o] * rsqrtf(var[o] + 1e-5f);
  const float mu = mean[o];
  const float bt = beta[o];
  float m = -3.4e38f;
  for (int k = 0; k < KNN_K; ++k) {
    const int idx = knn[n * KNN_K + k];
    const float y = U[(size_t)(b * NPTS + idx) * CDIM + o] + vv;
    float z = (y - mu) * sc + bt;
    z = (z >= 0.f) ? z : 0.2f * z;
    m = fmaxf(m, z);
  }
  Out[(size_t)bn * CDIM + o] = m;
}

// ---------------------------------------------------------------------------
// Kernel 6: final 64->2 head + transpose into out[0 .. B*2*N)
// ---------------------------------------------------------------------------
__global__ void bits_out(const float* __restrict__ Q2, const float* __restrict__ w3,
                         const float* __restrict__ b3, float* __restrict__ out) {
  const int t = blockIdx.x * blockDim.x + threadIdx.x;
  if (t >= BSZ * NPTS) return;
  const int b = t >> 9, n = t & 511;
  const float* q = Q2 + (size_t)t * 64;
  float s0 = b3[0], s1 = b3[1];
#pragma unroll 4
  for (int j = 0; j < 64; ++j) {
    const float qv = q[j];
    s0 += qv * w3[j * 2 + 0];
    s1 += qv * w3[j * 2 + 1];
  }
  out[(size_t)(b * 2 + 0) * NPTS + n] = s0;
  out[(size_t)(b * 2 + 1) * NPTS + n] = s1;
}

// ---------------------------------------------------------------------------
// Kernel 7: transpose h (B,N,256) -> out (B,256,N) at offset
// ---------------------------------------------------------------------------
__global__ __launch_bounds__(256) void transpose_h(const float* __restrict__ H,
                                                   float* __restrict__ out) {
  const int bn = blockIdx.x;
  const int n = bn & 511;
  const int b = bn >> 9;
  const int o = threadIdx.x;
  out[(size_t)(b * CDIM + o) * NPTS + n] = H[(size_t)bn * CDIM + o];
}

// ---------------------------------------------------------------------------
extern "C" void kernel_launch(void* const* d_in, const int* in_sizes, int n_in,
                              void* d_out, int out_size, void* d_ws,
                              size_t ws_size, hipStream_t stream) {
  (void)in_sizes; (void)n_in; (void)out_size; (void)ws_size;
  const float* img   = (const float*)d_in[0];
  const float* gfeat = (const float*)d_in[1];
  const float* roi   = (const float*)d_in[2];
  const int*   px    = (const int*)d_in[3];
  const int*   py    = (const int*)d_in[4];
  const int*   knn   = (const int*)d_in[5];
  const float* convw = (const float*)d_in[6];
  const float* convb = (const float*)d_in[7];
  const float* pw1   = (const float*)d_in[8];
  const float* pb1   = (const float*)d_in[9];
  const float* pw2   = (const float*)d_in[10];
  const float* pb2   = (const float*)d_in[11];
  const float* g1w   = (const float*)d_in[12];
  const float* g1g   = (const float*)d_in[13];
  const float* g1b   = (const float*)d_in[14];
  const float* g1m   = (const float*)d_in[15];
  const float* g1v   = (const float*)d_in[16];
  const float* g2w   = (const float*)d_in[17];
  const float* g2g   = (const float*)d_in[18];
  const float* g2b   = (const float*)d_in[19];
  const float* g2m   = (const float*)d_in[20];
  const float* g2v   = (const float*)d_in[21];
  const float* qw1   = (const float*)d_in[22];
  const float* qb1   = (const float*)d_in[23];
  const float* qw2   = (const float*)d_in[24];
  const float* qb2   = (const float*)d_in[25];
  const float* qw3   = (const float*)d_in[26];
  const float* qb3   = (const float*)d_in[27];
  float* out = (float*)d_out;
  float* ws = (float*)d_ws;

  const size_t M = (size_t)BSZ * NPTS;          // 8192
  float* local320 = ws;                         // 8192*320
  float* buf1 = local320 + M * 320;             // 8192*256
  float* buf2 = buf1 + M * 256;
  float* buf3 = buf2 + M * 256;
  float* buf4 = buf3 + M * 256;
  float* Wu1 = buf4 + M * 256;                  // 256*256 each
  float* Wv1 = Wu1 + 65536;
  float* Wu2 = Wv1 + 65536;
  float* Wv2 = Wu2 + 65536;

  // 1. gathered conv -> local320[:, 0:256]
  conv_gather_wmma<<<512, 256, 0, stream>>>(img, px, py, convw, convb, roi, local320);
  // 2. graph concat -> local320[:, 256:320]
  graph_concat<<<(BSZ * NPTS * GDIM_) / 256, 256, 0, stream>>>(gfeat, local320);
  // 3. EdgeConv weight prep
  edge_weight_prep<<<256, 256, 0, stream>>>(g1w, Wu1, Wv1);
  edge_weight_prep<<<256, 256, 0, stream>>>(g2w, Wu2, Wv2);
  // 4. pre MLP
  gemm_bias_lrelu<<<dim3(128, 8), 256, 0, stream>>>(local320, pw1, pb1, buf1, 320, 256, 0.01f);
  gemm_bias_lrelu<<<dim3(128, 8), 256, 0, stream>>>(buf1, pw2, pb2, buf2, 256, 256, 0.01f);
  // 5. EdgeConv 1: u = x@Wu1, v = x@Wv1, gather-max
  gemm_bias_lrelu<<<dim3(128, 8), 256, 0, stream>>>(buf2, Wu1, nullptr, buf1, 256, 256, 1.0f);
  gemm_bias_lrelu<<<dim3(128, 8), 256, 0, stream>>>(buf2, Wv1, nullptr, buf3, 256, 256, 1.0f);
  edge_gather_max<<<BSZ * NPTS, 256, 0, stream>>>(buf1, buf3, knn, g1g, g1b, g1m, g1v, buf4);
  // 6. EdgeConv 2
  gemm_bias_lrelu<<<dim3(128, 8), 256, 0, stream>>>(buf4, Wu2, nullptr, buf1, 256, 256, 1.0f);
  gemm_bias_lrelu<<<dim3(128, 8), 256, 0, stream>>>(buf4, Wv2, nullptr, buf3, 256, 256, 1.0f);
  edge_gather_max<<<BSZ * NPTS, 256, 0, stream>>>(buf1, buf3, knn, g2g, g2b, g2m, g2v, buf2);
  // 7. q head
  gemm_bias_lrelu<<<dim3(128, 8), 256, 0, stream>>>(buf2, qw1, qb1, buf1, 256, 256, 0.01f);
  gemm_bias_lrelu<<<dim3(128, 2), 256, 0, stream>>>(buf1, qw2, qb2, buf3, 256, 64, 0.01f);
  // 8. outputs: bits (B,2,N) then h (B,256,N)
  bits_out<<<32, 256, 0, stream>>>(buf3, qw3, qb3, out);
  transpose_h<<<BSZ * NPTS, 256, 0, stream>>>(buf2, out + (size_t)BSZ * 2 * NPTS);
}